// SisterRegressionROIPooling_65180423684401
// MI455X (gfx1250) — compile-verified
//
#include <hip/hip_runtime.h>

// ---------------------------------------------------------------------------
// SisterRegressionROIPooling for MI455X (gfx1250, wave32).
//   Phase A: row-inclusive prefix sums via V_WMMA_F32_16X16X4_F32 (P = X * U),
//            tiles staged coalesced into LDS with async-to-LDS DMA (ASYNCcnt),
//            double-buffered to overlap DMA with the serial WMMA carry chain.
//   Phase B: col-inclusive prefix sums in-place via WMMA (P = L * X), with
//            global_prefetch of the next tile down the column.
//   Phase C: per-ROI bilinear form u^T G v over SAT corner samples (L2 gathers)
// ---------------------------------------------------------------------------

#define HH 1024
#define WW 1024
#define NCH 8
#define CH_STRIDE ((size_t)HH * WW)

typedef float v2f __attribute__((ext_vector_type(2)));
typedef float v8f __attribute__((ext_vector_type(8)));

#if defined(__gfx1250__) &&                                                   \
    __has_builtin(__builtin_amdgcn_global_load_async_to_lds_b128) &&          \
    __has_builtin(__builtin_amdgcn_s_wait_asynccnt)
#define USE_ASYNC_LDS 1
// Parameter types per hipcc diagnostic: int __attribute__((vector_size(16)))*
typedef int vsi4 __attribute__((vector_size(16)));
#define GPTR4(p) ((__attribute__((address_space(1))) vsi4*)(p))
#define LPTR4(p) ((__attribute__((address_space(3))) vsi4*)(p))
#else
#define USE_ASYNC_LDS 0
#endif

// ---------------------------------------------------------------------------
// Phase A: row scan. One wave per (channel, 16-row strip). 512 waves total.
// Tile prefix P = X * U via 4 chained 16x16x4 f32 WMMAs; serial carry per row.
// ---------------------------------------------------------------------------
__global__ __launch_bounds__(256) void sat_rowscan_kernel(
    const float* __restrict__ in, float* __restrict__ sat) {
  const int lane = threadIdx.x & 31;
  const int wid  = threadIdx.x >> 5;
  const int s    = blockIdx.x * 8 + wid;        // 0..511
  const int c    = s >> 6;
  const int y0   = (s & 63) << 4;
  const int hi   = lane >> 4;                   // lane half (K-split / M-split)
  const int m    = lane & 15;                   // A-matrix row / D-matrix col

  const float* src = in  + (size_t)c * CH_STRIDE + (size_t)y0 * WW;
  float*       dst = sat + (size_t)c * CH_STRIDE + (size_t)y0 * WW;

#if USE_ASYNC_LDS
  // Double-buffered 16x16 tile per wave: 8 waves * 2 * 1KB = 16KB LDS.
  __shared__ float tileBuf[8][2][16 * 16];
  float* const tb0 = &tileBuf[wid][0][0];
  float* const tb1 = &tileBuf[wid][1][0];
  const int arow = lane >> 2;                   // 0..7   (staging row)
  const int acol = (lane & 3) << 2;             // 0,4,8,12 (staging col, b128)
  // Coalesced DMA of one 16x16 tile into LDS: 2 x b128 per lane.
  auto stage = [&](int t, float* tb) {
    const float* g0 = src + (size_t)arow * WW + (t << 4) + acol;
    __builtin_amdgcn_global_load_async_to_lds_b128(
        GPTR4(g0), LPTR4(tb + arow * 16 + acol), 0, 0);
    __builtin_amdgcn_global_load_async_to_lds_b128(
        GPTR4(g0 + (size_t)8 * WW), LPTR4(tb + (arow + 8) * 16 + acol), 0, 0);
  };
  stage(0, tb0);
#endif

  float carry[8];
#pragma unroll
  for (int i = 0; i < 8; ++i) carry[i] = 0.f;

  for (int t = 0; t < 64; ++t) {
    const int x0 = t << 4;
#if USE_ASYNC_LDS
    float* const tb = (t & 1) ? tb1 : tb0;
    __builtin_amdgcn_s_wait_asynccnt(0);        // tile t resident in LDS
    if (t < 63) stage(t + 1, (t & 1) ? tb0 : tb1);  // overlap next DMA
#endif
    v8f acc = {};
#pragma unroll
    for (int kb = 0; kb < 4; ++kb) {
      const int k0 = kb * 4 + 2 * hi;           // this lane's K pair
      // A layout: lane holds X[row m][k0], X[row m][k0+1]
      v2f a;
#if USE_ASYNC_LDS
      a = *(const v2f*)(tb + m * 16 + k0);      // ds_load_b64, 8B aligned
#else
      const float* ap = src + (size_t)m * WW + x0 + k0;
      a.x = ap[0]; a.y = ap[1];
#endif
      // B = U slice: U[k][n] = (k <= n), n = m
      v2f b;
      b.x = (k0     <= m) ? 1.f : 0.f;
      b.y = (k0 + 1 <= m) ? 1.f : 0.f;
      acc = __builtin_amdgcn_wmma_f32_16x16x4_f32(
          false, a, false, b, (short)0, acc, false, false);
    }
    // D layout: acc[i] = P[row i + 8*hi][col m]. Add carry, refresh carry from
    // column 15 (lane 15 holds rows 0..7, lane 31 holds rows 8..15).
#pragma unroll
    for (int i = 0; i < 8; ++i) {
      float o = acc[i] + carry[i];
      carry[i] = __shfl(o, lane | 15, 32);
      dst[(size_t)(i + 8 * hi) * WW + x0 + m] = o;
    }
  }
}

// ---------------------------------------------------------------------------
// Phase B: column scan, in place. One wave per (channel, 16-col band).
// Tile prefix P = L * X; serial carry per column.
// ---------------------------------------------------------------------------
__global__ __launch_bounds__(256) void sat_colscan_kernel(float* __restrict__ sat) {
  const int lane = threadIdx.x & 31;
  const int wid  = threadIdx.x >> 5;
  const int s    = blockIdx.x * 8 + wid;        // 0..511
  const int c    = s >> 6;
  const int x0   = (s & 63) << 4;
  const int hi   = lane >> 4;
  const int n    = lane & 15;                   // A row (M) and B/D col (N)

  float* base = sat + (size_t)c * CH_STRIDE;
  float carry = 0.f;

  for (int t = 0; t < 64; ++t) {
    const int y0 = t << 4;
    // Warm L2/L0 for the next tile down the column (global_prefetch_b8).
    if (t < 63) {
      const float* pf = base + (size_t)(y0 + 16 + n) * WW + x0 + (hi << 3);
      __builtin_prefetch((const void*)pf, 0, 1);
    }
    v8f acc = {};
#pragma unroll
    for (int kb = 0; kb < 4; ++kb) {
      const int k0 = kb * 4 + 2 * hi;
      // A = L slice: L[M][k] = (k <= M), M = n
      v2f a;
      a.x = (k0     <= n) ? 1.f : 0.f;
      a.y = (k0 + 1 <= n) ? 1.f : 0.f;
      // B = X rows k0, k0+1 at column n (coalesced across lanes 0-15 / 16-31)
      v2f b;
      b.x = base[(size_t)(y0 + k0    ) * WW + x0 + n];
      b.y = base[(size_t)(y0 + k0 + 1) * WW + x0 + n];
      acc = __builtin_amdgcn_wmma_f32_16x16x4_f32(
          false, a, false, b, (short)0, acc, false, false);
    }
    float outv[8];
#pragma unroll
    for (int i = 0; i < 8; ++i) outv[i] = acc[i] + carry;
    // column sum of tile = P[15][n], held by lane 16+n in acc slot 7
    carry = __shfl(outv[7], 16 + n, 32);
#pragma unroll
    for (int i = 0; i < 8; ++i)
      base[(size_t)(y0 + i + 8 * hi) * WW + x0 + n] = outv[i];
  }
}

// ---------------------------------------------------------------------------
// Phase C: ROI pooling. One wave per ROI.
//   out[c] = (1/49) * sum_{p,q} w_y[p] * w_x[q] * SAT[c][yb[p]][xb[q]]
// lane half -> even/odd p in 0..13, lane&15 -> q in 0..13 (q>=14 idle).
// SAT stored border-free: SAT(y,x) = buf[y-1][x-1], zero if y==0 or x==0.
// ---------------------------------------------------------------------------
__global__ __launch_bounds__(256) void roipool_bilinear_kernel(
    const int* __restrict__ rois, const float* __restrict__ sat,
    float* __restrict__ out) {
  const int lane = threadIdx.x & 31;
  const int wid  = threadIdx.x >> 5;
  const int r    = blockIdx.x * 8 + wid;        // 0..8191
  const int h    = lane >> 4;
  const int q    = lane & 15;

  const int4 rv = ((const int4*)rois)[r];
  const int yA = rv.x >> 5;                     // // FEAT_STRIDE (nonneg)
  const int xA = rv.y >> 5;
  const int yB = (rv.z >> 5) + 1;               // exclusive ends
  const int xB = (rv.w >> 5) + 1;
  const int leny = yB - yA;
  const int lenx = xB - xA;

  // this lane's x boundary + signed weight
  int   xb = 0;
  float wx = 0.f;
  if (q < 14) {
    const int j  = (q < 7) ? q : q - 7;
    const int lo = xA + (j * lenx) / 7;
    const int hx = xA + ((j + 1) * lenx + 6) / 7;
    const float inv = 1.f / (float)(hx - lo);
    if (q < 7) { xb = lo; wx = -inv; } else { xb = hx; wx = inv; }
  }

  float acc[NCH];
#pragma unroll
  for (int c = 0; c < NCH; ++c) acc[c] = 0.f;

  if (wx != 0.f) {
    for (int t = 0; t < 7; ++t) {
      const int p  = 2 * t + h;                 // 0..13, disjoint per half
      const int j  = (p < 7) ? p : p - 7;
      const int lo = yA + (j * leny) / 7;
      const int hy = yA + ((j + 1) * leny + 6) / 7;
      const float inv = 1.f / (float)(hy - lo);
      int   yb; float wy;
      if (p < 7) { yb = lo; wy = -inv; } else { yb = hy; wy = inv; }
      if (yb > 0 && xb > 0) {
        const float wyx = wy * wx;
        const float* g = sat + (size_t)(yb - 1) * WW + (xb - 1);
#pragma unroll
        for (int c = 0; c < NCH; ++c)
          acc[c] += wyx * g[(size_t)c * CH_STRIDE];
      }
    }
  }

#pragma unroll
  for (int c = 0; c < NCH; ++c) {
    float v = acc[c];
    for (int off = 16; off; off >>= 1) v += __shfl_xor(v, off, 32);
    if (lane == 0) out[(size_t)r * NCH + c] = v * (1.f / 49.f);
  }
}

// ---------------------------------------------------------------------------
extern "C" void kernel_launch(void* const* d_in, const int* in_sizes, int n_in,
                              void* d_out, int out_size, void* d_ws, size_t ws_size,
                              hipStream_t stream) {
  const float* conv = (const float*)d_in[0];   // [1,8,1024,1024] f32
  const int*   rois = (const int*)d_in[1];     // [8192,4] int
  float* sat = (float*)d_ws;                   // 8*1024*1024 f32 = 32 MiB
  float* out = (float*)d_out;                  // [8192,2,4] f32

  sat_rowscan_kernel<<<64, 256, 0, stream>>>(conv, sat);
  sat_colscan_kernel<<<64, 256, 0, stream>>>(sat);
  roipool_bilinear_kernel<<<1024, 256, 0, stream>>>(rois, sat, out);
}